// DecoupledGCNUnit_7885559955768
// MI455X (gfx1250) — compile-verified
//
#include <hip/hip_runtime.h>
#include <hip/hip_bf16.h>

// Problem dims
#define NN   64
#define CIN  64
#define COUT 64
#define TT   256
#define VV   25
#define KK   3
#define GG   8
#define DOUT 192          // C_OUT*K
#define TV   6400         // T*V
#define VPAD 28           // V padded to multiple of 4 (WMMA K-step)
#define WPAD 32           // w padded to 2x16 WMMA N-tiles
#define MCNT 409600.0f    // N*T*V, per-channel BN element count

typedef float v2f __attribute__((ext_vector_type(2)));
typedef float v8f __attribute__((ext_vector_type(8)));

static __device__ __forceinline__ v8f wmma_f32_4(v2f a, v2f b, v8f c) {
#if __has_builtin(__builtin_amdgcn_wmma_f32_16x16x4_f32)
  // D(16x16,f32) = A(16x4,f32) x B(4x16,f32) + C
  return __builtin_amdgcn_wmma_f32_16x16x4_f32(false, a, false, b, (short)0, c,
                                               false, false);
#else
  c[0] += a.x * b.x + a.y * b.y;  // fallback only so compile never fails
  return c;
#endif
}

// ---------------------------------------------------------------------------
// k_stats: one pass over x0 computing per-channel sums s[64] and the Gram
// matrix G[64][64] = sum_m x0[:,m] x0[:,m]^T via WMMA.
// grid = (5 tv-groups, 64 n), block 256 (8 waves). Each block: 10 chunks of
// 128 tv staged in LDS; each wave owns 2 of the 16 (c1,c2) 16x16 Gram tiles,
// accumulating in registers across chunks; one atomicAdd per element at end.
// ---------------------------------------------------------------------------
__global__ __launch_bounds__(256) void k_stats(const float* __restrict__ x0,
                                               float* __restrict__ G,
                                               float* __restrict__ s) {
  __shared__ __align__(16) float sX[CIN][128];
  __shared__ float sRed[256];
  const int tid = threadIdx.x, wave = tid >> 5, lane = tid & 31;
  const int col = lane & 15, khalf = (lane >> 4) << 1, rowoff = (lane >> 4) << 3;
  const int n = blockIdx.y;
  const float* xn = x0 + (size_t)n * CIN * TV + (size_t)blockIdx.x * 1280;

  const int t0 = wave * 2, t1 = wave * 2 + 1;       // this wave's Gram tiles
  const int i0 = t0 >> 2, j0 = t0 & 3, i1 = t1 >> 2, j1 = t1 & 3;
  v8f g0 = {}, g1 = {};
  float csum = 0.f;
  const int myc = tid >> 2, seg = (tid & 3) << 5;

  for (int ch = 0; ch < 10; ++ch) {
    __syncthreads();
    for (int i = tid; i < CIN * 32; i += 256) {
      const int c = i >> 5, j4 = (i & 31) << 2;
      *(float4*)&sX[c][j4] = *(const float4*)(xn + (size_t)c * TV + ch * 128 + j4);
    }
    __syncthreads();
    for (int u = 0; u < 32; ++u) csum += sX[myc][seg + u];
    for (int ms = 0; ms < 128; ms += 4) {
      v2f a, b;
      a.x = sX[i0 * 16 + col][ms + khalf];  a.y = sX[i0 * 16 + col][ms + khalf + 1];
      b.x = sX[j0 * 16 + col][ms + khalf];  b.y = sX[j0 * 16 + col][ms + khalf + 1];
      g0 = wmma_f32_4(a, b, g0);
      a.x = sX[i1 * 16 + col][ms + khalf];  a.y = sX[i1 * 16 + col][ms + khalf + 1];
      b.x = sX[j1 * 16 + col][ms + khalf];  b.y = sX[j1 * 16 + col][ms + khalf + 1];
      g1 = wmma_f32_4(a, b, g1);
    }
  }
  for (int r = 0; r < 8; ++r) {
    atomicAdd(&G[(i0 * 16 + r + rowoff) * CIN + j0 * 16 + col], g0[r]);
    atomicAdd(&G[(i1 * 16 + r + rowoff) * CIN + j1 * 16 + col], g1[r]);
  }
  sRed[tid] = csum;
  __syncthreads();
  if (tid < CIN)
    atomicAdd(&s[tid], sRed[tid * 4] + sRed[tid * 4 + 1] +
                       sRed[tid * 4 + 2] + sRed[tid * 4 + 3]);
}

// ---------------------------------------------------------------------------
// k_prep (1 block): normalized padded adjacency nA; analytic BN0 stats from
// (s, G): scale0/shift0; scaled weights SW = W*diag(scale0); per-(c,w)
// constant cshift folding shift0 through the adjacency.
// ---------------------------------------------------------------------------
__global__ __launch_bounds__(256) void k_prep(
    const float* __restrict__ dA, const float* __restrict__ W,
    const float* __restrict__ bias, const float* __restrict__ g0v,
    const float* __restrict__ b0v, const float* __restrict__ G,
    const float* __restrict__ s, float* __restrict__ nA,
    float* __restrict__ SW, float* __restrict__ scale0,
    float* __restrict__ shift0, float* __restrict__ cshift) {
  const int tid = threadIdx.x;
  // 1) degree-normalized adjacency, zero-padded to [K][G][VPAD][WPAD]
  for (int idx = tid; idx < KK * GG * WPAD; idx += 256) {
    const int kg = idx >> 5, w = idx & 31;
    const float* src = dA + (size_t)kg * VV * VV;
    float* dst = nA + (size_t)kg * VPAD * WPAD;
    float inv = 0.f;
    if (w < VV) {
      float t = 0.f;
      for (int v = 0; v < VV; ++v) t += src[v * VV + w];
      inv = 1.f / (t + 0.001f);
    }
    for (int v = 0; v < VPAD; ++v)
      dst[v * WPAD + w] = (v < VV && w < VV) ? src[v * VV + w] * inv : 0.f;
  }
  // 2) analytic BN0: y_d = x0.W_d + b  =>  E[y], E[y^2] from s and G
  if (tid < DOUT) {
    const int d = tid;
    float sy = 0.f;
    for (int c = 0; c < CIN; ++c) sy += s[c] * W[c * DOUT + d];
    float q = 0.f;
    for (int c1 = 0; c1 < CIN; ++c1) {
      float t = 0.f;
      const float* Gr = G + c1 * CIN;
      for (int c2 = 0; c2 < CIN; ++c2) t += Gr[c2] * W[c2 * DOUT + d];
      q += W[c1 * DOUT + d] * t;
    }
    const float b = bias[d];
    const float mean = (sy + MCNT * b) * (1.f / MCNT);
    const float sumsq = q + 2.f * b * sy + MCNT * b * b;
    const float var = sumsq * (1.f / MCNT) - mean * mean;
    const float sc = g0v[d] * rsqrtf(var + 1e-5f);
    scale0[d] = sc;
    shift0[d] = b0v[d] - mean * sc;
  }
  __threadfence_block();
  __syncthreads();
  // 3) fold scale0 into the weights
  for (int idx = tid; idx < CIN * DOUT; idx += 256)
    SW[idx] = W[idx] * scale0[idx % DOUT];
  // 4) fold shift0 through adjacency column sums into cshift[c][w]
  for (int i = tid; i < COUT * WPAD; i += 256) {
    const int c = i >> 5, w = i & 31, g = c >> 3;
    float val = 0.f;
    if (w < VV) {
      for (int k = 0; k < KK; ++k) {
        const float* p = nA + (size_t)(k * GG + g) * VPAD * WPAD + w;
        float S = 0.f;
        for (int v = 0; v < VV; ++v) S += p[v * WPAD];
        val += shift0[k * COUT + c] * S;
      }
    }
    cshift[i] = val;
  }
}

// ---------------------------------------------------------------------------
// k_fused: per (n, 16-t chunk) block, rotate over k-subsets:
//   phase A: y_s[64d][16t][25v] = x0 . SW_k   (WMMA, result into LDS)
//   phase B: z-accumulators += y_s x nA_k     (WMMA, 8 channels per wave)
// Epilogue: z = acc + cshift, fused BN1 stats. y never touches HBM.
// grid = (16 tchunk, 64 n), block 256 (8 waves). Dynamic LDS 160,256 B.
// ---------------------------------------------------------------------------
__global__ __launch_bounds__(256) void k_fused(
    const float* __restrict__ x0, const float* __restrict__ SW,
    const float* __restrict__ nA, const float* __restrict__ cshift,
    float* __restrict__ z, float* __restrict__ sum1, float* __restrict__ sq1) {
  extern __shared__ float smem[];
  float* sY  = smem;                 // [64][16][VPAD] = 28672 (v>=25 stays 0)
  float* sSW = sY + 64 * 16 * VPAD;  // [64][64] = 4096
  float* sNA = sSW + 4096;           // [8][VPAD][WPAD] = 7168
  float* sCs = sNA + 7168;           // [64]
  float* sCq = sCs + 64;             // [64]

  const int tid = threadIdx.x, wave = tid >> 5, lane = tid & 31;
  const int col = lane & 15, khalf = (lane >> 4) << 1, rowoff = (lane >> 4) << 3;
  const int n = blockIdx.y, tbase = blockIdx.x * 16;
  const float* xn = x0 + (size_t)n * CIN * TV + (size_t)tbase * VV;

  for (int i = tid; i < 64 * 16 * VPAD; i += 256) sY[i] = 0.f;  // zero pads
  if (tid < 128) sCs[tid] = 0.f;                                 // sCs+sCq

  v8f acc0[8] = {}, acc1[8] = {};

  for (int k = 0; k < KK; ++k) {
    __syncthreads();  // previous phase B done before overwriting sSW/sNA/sY
    for (int i = tid; i < 4096; i += 256) {
      const int c = i >> 6, dl = i & 63;
      sSW[i] = SW[c * DOUT + k * COUT + dl];
    }
    for (int i = tid; i < GG * VPAD * WPAD; i += 256)
      sNA[i] = nA[(size_t)k * GG * VPAD * WPAD + i];
    __syncthreads();

    // ---- phase A: wave owns tv-tiles tt, tt+8, ... (25 tiles of 16 tv)
    for (int tt = wave; tt < 25; tt += 8) {
      const int tvp = tt * 16 + col;
      v8f ya[4] = {};
      for (int cs = 0; cs < CIN; cs += 4) {
        v2f b;
        b.x = xn[(size_t)(cs + khalf) * TV + tvp];
        b.y = xn[(size_t)(cs + khalf + 1) * TV + tvp];
#pragma unroll
        for (int dt = 0; dt < 4; ++dt) {
          v2f a;
          a.x = sSW[(cs + khalf) * 64 + dt * 16 + col];
          a.y = sSW[(cs + khalf + 1) * 64 + dt * 16 + col];
          ya[dt] = wmma_f32_4(a, b, ya[dt]);
        }
      }
      const int tl = tvp / VV, vv = tvp - tl * VV;
      const int toff = tl * VPAD + vv;
#pragma unroll
      for (int dt = 0; dt < 4; ++dt)
#pragma unroll
        for (int r = 0; r < 8; ++r)
          sY[(dt * 16 + r + rowoff) * (16 * VPAD) + toff] = ya[dt][r];
    }
    __syncthreads();

    // ---- phase B: wave handles c in [wave*8, wave*8+8); g == wave
    const float* bb = sNA + wave * VPAD * WPAD;
#pragma unroll
    for (int j = 0; j < 8; ++j) {
      const int c = wave * 8 + j;
      const float* ar = sY + c * (16 * VPAD) + (lane & 15) * VPAD;
      for (int vs = 0; vs < VPAD; vs += 4) {
        v2f a, b0, b1;
        a.x = ar[vs + khalf];
        a.y = ar[vs + khalf + 1];
        b0.x = bb[(vs + khalf) * WPAD + col];
        b0.y = bb[(vs + khalf + 1) * WPAD + col];
        b1.x = bb[(vs + khalf) * WPAD + 16 + col];
        b1.y = bb[(vs + khalf + 1) * WPAD + 16 + col];
        acc0[j] = wmma_f32_4(a, b0, acc0[j]);
        acc1[j] = wmma_f32_4(a, b1, acc1[j]);
      }
    }
  }

  // ---- epilogue: cshift add, store z, fused BN1 stats
#pragma unroll
  for (int j = 0; j < 8; ++j) {
    const int c = wave * 8 + j;
    float lsum = 0.f, lsq = 0.f;
    float* zr = z + (((size_t)n * COUT + c) * TT + tbase) * VV;
    const float cs0 = cshift[c * WPAD + col];
    const bool hi = (16 + col) < VV;
    const float cs1 = hi ? cshift[c * WPAD + 16 + col] : 0.f;
#pragma unroll
    for (int r = 0; r < 8; ++r) {
      const int tl = r + rowoff;
      const float v0 = acc0[j][r] + cs0;
      zr[(size_t)tl * VV + col] = v0;
      lsum += v0; lsq += v0 * v0;
      if (hi) {
        const float v1 = acc1[j][r] + cs1;
        zr[(size_t)tl * VV + 16 + col] = v1;
        lsum += v1; lsq += v1 * v1;
      }
    }
    atomicAdd(&sCs[c], lsum);
    atomicAdd(&sCq[c], lsq);
  }
  __syncthreads();
  if (tid < COUT) {
    atomicAdd(&sum1[tid], sCs[tid]);
    atomicAdd(&sq1[tid], sCq[tid]);
  }
}

// ---------------------------------------------------------------------------
__global__ void k_bn1(const float* __restrict__ sum1,
                      const float* __restrict__ sq1,
                      const float* __restrict__ g1,
                      const float* __restrict__ b1,
                      float* __restrict__ scale1, float* __restrict__ shift1) {
  const int c = threadIdx.x;
  if (c < COUT) {
    const float mean = sum1[c] * (1.0f / MCNT);
    const float var  = sq1[c] * (1.0f / MCNT) - mean * mean;
    const float sc   = g1[c] * rsqrtf(var + 1e-5f);
    scale1[c] = sc;
    shift1[c] = b1[c] - mean * sc;
  }
}

// ---------------------------------------------------------------------------
// k_final: out = relu(scale1[c]*z + shift1[c] + x0); z (105 MB) is L2-resident
// ---------------------------------------------------------------------------
__global__ __launch_bounds__(256) void k_final(
    const float* __restrict__ z, const float* __restrict__ x0,
    const float* __restrict__ scale1, const float* __restrict__ shift1,
    float* __restrict__ out, int n4) {
  const int i = blockIdx.x * 256 + threadIdx.x;
  if (i >= n4) return;
  const int c = ((i << 2) / TV) & (COUT - 1);
  const float4 zv = ((const float4*)z)[i];
  const float4 xv = ((const float4*)x0)[i];
  const float sc = scale1[c], sh = shift1[c];
  float4 o;
  o.x = fmaxf(zv.x * sc + sh + xv.x, 0.f);
  o.y = fmaxf(zv.y * sc + sh + xv.y, 0.f);
  o.z = fmaxf(zv.z * sc + sh + xv.z, 0.f);
  o.w = fmaxf(zv.w * sc + sh + xv.w, 0.f);
  ((float4*)out)[i] = o;
}

// ---------------------------------------------------------------------------
extern "C" void kernel_launch(void* const* d_in, const int* in_sizes, int n_in,
                              void* d_out, int out_size, void* d_ws,
                              size_t ws_size, hipStream_t stream) {
  const float* x0   = (const float*)d_in[0];   // [64,64,256,25]
  const float* dA   = (const float*)d_in[1];   // [3,8,25,25]
  const float* W    = (const float*)d_in[2];   // [64,192]
  const float* bias = (const float*)d_in[3];   // [192]
  const float* g0   = (const float*)d_in[4];   // [192]
  const float* b0   = (const float*)d_in[5];   // [192]
  const float* g1   = (const float*)d_in[6];   // [64]
  const float* b1   = (const float*)d_in[7];   // [64]
  float* out = (float*)d_out;

  float* ws     = (float*)d_ws;
  float* z      = ws;                                   // 26,214,400
  float* nA     = z + (size_t)NN * COUT * TV;           // 21,504
  float* cshift = nA + (size_t)KK * GG * VPAD * WPAD;   // 2,048
  float* SW     = cshift + (size_t)COUT * WPAD;         // 12,288
  float* zr0    = SW + (size_t)CIN * DOUT;              // zeroed region:
  float* G      = zr0;                                  //   4096
  float* s      = G + CIN * CIN;                        //   64
  float* sum1   = s + CIN;                              //   64
  float* sq1    = sum1 + COUT;                          //   64  (=4288 total)
  float* scale0 = zr0 + 4288;                           // 192
  float* shift0 = scale0 + DOUT;                        // 192
  float* scale1 = shift0 + DOUT;                        // 64
  float* shift1 = scale1 + COUT;                        // 64

  hipMemsetAsync(zr0, 0, 4288 * sizeof(float), stream);

  k_stats<<<dim3(5, NN), 256, 0, stream>>>(x0, G, s);
  k_prep<<<1, 256, 0, stream>>>(dA, W, bias, g0, b0, G, s, nA, SW, scale0,
                                shift0, cshift);
  const size_t lds_bytes = (size_t)(64 * 16 * VPAD + 4096 + 7168 + 128) * 4;
  k_fused<<<dim3(16, NN), 256, lds_bytes, stream>>>(x0, SW, nA, cshift, z,
                                                    sum1, sq1);
  k_bn1<<<1, 64, 0, stream>>>(sum1, sq1, g1, b1, scale1, shift1);
  const int n4 = (NN * COUT * TT * VV) / 4;   // 6,553,600
  k_final<<<(n4 + 255) / 256, 256, 0, stream>>>(z, x0, scale1, shift1, out, n4);
}